// HeteroGAT_12884901888073
// MI455X (gfx1250) — compile-verified
//
#include <hip/hip_runtime.h>
#include <math.h>

#define FC 128
#define NEG_SLOPE 0.2f
#define LN_EPS 1e-5f

typedef float v2f __attribute__((ext_vector_type(2)));
typedef float v8f __attribute__((ext_vector_type(8)));

// ---------------- float atomic max via int bit trick (lattice-correct) -----
__device__ __forceinline__ void atomicMaxF(float* addr, float val) {
    if (val >= 0.f)
        atomicMax((int*)addr, __float_as_int(val));
    else
        atomicMin((unsigned int*)addr, __float_as_uint(val));
}

// ---------------- fill ------------------------------------------------------
__global__ void fill_f32(float* __restrict__ p, float v, size_t n) {
    size_t i = (size_t)blockIdx.x * blockDim.x + threadIdx.x;
    if (i < n) p[i] = v;
}

// ---------------- wtil = W @ att (128-vector) -------------------------------
__global__ void compute_wtil(const float* __restrict__ W, const float* __restrict__ att,
                             float* __restrict__ wtil) {
    int i = threadIdx.x;  // 128 threads
    float s = 0.f;
    for (int j = 0; j < FC; ++j) s += W[i * FC + j] * att[j];
    wtil[i] = s;
}

// ---------------- a[row] = dot(X[row], wtil)  (one wave32 per row) ----------
__global__ void matvec_att(const float* __restrict__ X, const float* __restrict__ wtil,
                           float* __restrict__ a, int n) {
    int lane = threadIdx.x & 31;
    int row  = (blockIdx.x * blockDim.x + threadIdx.x) >> 5;
    if (row >= n) return;
    float4 x = *(const float4*)(X + (size_t)row * FC + lane * 4);
    float4 w = *(const float4*)(wtil + lane * 4);
    float s = x.x * w.x + x.y * w.y + x.z * w.z + x.w * w.w;
    #pragma unroll
    for (int m = 16; m >= 1; m >>= 1) s += __shfl_xor(s, m, 32);
    if (lane == 0) a[row] = s;
}

// ---------------- H[n x 128] = X[n x 128] @ W[128 x 128]  (fp32 WMMA) -------
// One wave computes a 16-row x 128-col strip via 8 accumulators.
// A(16x4 f32): lane l -> row (l&15); VGPR0/1 = K 2*(l>>4), 2*(l>>4)+1.
// B(4x16 f32): lane l -> col (l&15); VGPR0/1 = K rows 2*(l>>4), 2*(l>>4)+1.
// C/D(16x16 f32): vgpr r, lane l -> row r+8*(l>>4), col (l&15).
__global__ __launch_bounds__(256) void gemm_wmma(const float* __restrict__ X,
                                                 const float* __restrict__ W,
                                                 float* __restrict__ H, int n) {
    __shared__ float sW[FC * FC];                     // 64 KB of the 320 KB/WGP LDS
    for (int i = threadIdx.x; i < FC * FC / 4; i += 256)
        ((float4*)sW)[i] = ((const float4*)W)[i];
    __syncthreads();

    int wave = threadIdx.x >> 5, lane = threadIdx.x & 31;
    int rowBase = (blockIdx.x * 8 + wave) * 16;
    if (rowBase >= n) return;

    int m  = lane & 15;
    int kh = lane >> 4;
    int arow = rowBase + m; if (arow >= n) arow = n - 1;   // clamp (stores masked)
    const float* xr = X + (size_t)arow * FC;

    v8f acc[8] = {};
    for (int k = 0; k < FC; k += 4) {
        v2f a;
        a.x = xr[k + 2 * kh];
        a.y = xr[k + 2 * kh + 1];
        #pragma unroll
        for (int nb = 0; nb < 8; ++nb) {
            v2f b;
            b.x = sW[(k + 2 * kh) * FC + nb * 16 + m];
            b.y = sW[(k + 2 * kh + 1) * FC + nb * 16 + m];
            acc[nb] = __builtin_amdgcn_wmma_f32_16x16x4_f32(
                false, a, false, b, (short)0, acc[nb], false, false);
        }
    }
    #pragma unroll
    for (int r = 0; r < 8; ++r) {
        int rr = rowBase + r + 8 * kh;
        if (rr < n) {
            #pragma unroll
            for (int nb = 0; nb < 8; ++nb)
                H[(size_t)rr * FC + nb * 16 + m] = acc[nb][r];
        }
    }
}

// ---------------- edge pass 1: leaky-relu logits, segment max, degree -------
__global__ void edge_logits_max(const int* __restrict__ src, const int* __restrict__ dst,
                                const float* __restrict__ a_src, const float* __restrict__ a_dst,
                                float* __restrict__ logits, float* __restrict__ m,
                                float* __restrict__ deg, int E) {
    int e = blockIdx.x * blockDim.x + threadIdx.x;
    if (e >= E) return;
    int s = src[e], d = dst[e];
    float l = a_src[s] + a_dst[d];
    l = (l >= 0.f) ? l : NEG_SLOPE * l;
    logits[e] = l;
    atomicMaxF(&m[d], l);
    atomicAdd(&deg[d], 1.f);
}

// ---------------- edge pass 2: e = exp(logit - m[d]); denom += e ------------
__global__ void edge_exp_denom(const int* __restrict__ dst, const float* __restrict__ m,
                               float* __restrict__ logits /*in logit / out e*/,
                               float* __restrict__ denom, int E) {
    int e = blockIdx.x * blockDim.x + threadIdx.x;
    if (e >= E) return;
    int d = dst[e];
    float md = m[d];
    if (!isfinite(md)) md = 0.f;           // matches reference's finite-guard
    float ev = expf(logits[e] - md);
    logits[e] = ev;
    atomicAdd(&denom[d], ev);
}

// ---------------- edge pass 3: agg[d] += alpha * h_src[s]  (wave per edge) --
__global__ void edge_scatter(const int* __restrict__ src, const int* __restrict__ dst,
                             const float* __restrict__ eval, const float* __restrict__ denom,
                             const float* __restrict__ H, float* __restrict__ out, int E) {
    int lane = threadIdx.x & 31;
    int e = (blockIdx.x * blockDim.x + threadIdx.x) >> 5;
    if (e >= E) return;
    int s = src[e], d = dst[e];
    float alpha = eval[e] / (denom[d] + 1e-16f);
    float4 h = *(const float4*)(H + (size_t)s * FC + lane * 4);
    float* o = out + (size_t)d * FC + lane * 4;
    atomicAdd(o + 0, alpha * h.x);
    atomicAdd(o + 1, alpha * h.y);
    atomicAdd(o + 2, alpha * h.z);
    atomicAdd(o + 3, alpha * h.w);
}

// ---------------- mean aggregation + bias -----------------------------------
__global__ void finalize_mean_bias(float* __restrict__ out, const float* __restrict__ deg,
                                   const float* __restrict__ bias, int n) {
    int i = blockIdx.x * blockDim.x + threadIdx.x;
    if (i >= n * FC) return;
    int row = i >> 7, c = i & 127;
    out[i] = out[i] / fmaxf(deg[row], 1.f) + bias[c];
}

// ---------------- LayerNorm + ReLU in place (wave per row) ------------------
__global__ void ln_relu(float* __restrict__ x, const float* __restrict__ w,
                        const float* __restrict__ b, int n) {
    int lane = threadIdx.x & 31;
    int row  = (blockIdx.x * blockDim.x + threadIdx.x) >> 5;
    if (row >= n) return;
    float4 v = *(const float4*)(x + (size_t)row * FC + lane * 4);
    float s  = v.x + v.y + v.z + v.w;
    float sq = v.x * v.x + v.y * v.y + v.z * v.z + v.w * v.w;
    #pragma unroll
    for (int m = 16; m >= 1; m >>= 1) { s += __shfl_xor(s, m, 32); sq += __shfl_xor(sq, m, 32); }
    float mu   = s * (1.f / FC);
    float var  = sq * (1.f / FC) - mu * mu;
    float rstd = rsqrtf(var + LN_EPS);
    float4 wv = *(const float4*)(w + lane * 4);
    float4 bv = *(const float4*)(b + lane * 4);
    float4 o;
    o.x = fmaxf((v.x - mu) * rstd * wv.x + bv.x, 0.f);
    o.y = fmaxf((v.y - mu) * rstd * wv.y + bv.y, 0.f);
    o.z = fmaxf((v.z - mu) * rstd * wv.z + bv.z, 0.f);
    o.w = fmaxf((v.w - mu) * rstd * wv.w + bv.w, 0.f);
    *(float4*)(x + (size_t)row * FC + lane * 4) = o;
}

// ---------------- host-side conv driver -------------------------------------
struct ConvBufs { float *H, *a_src, *a_dst, *wtil_s, *wtil_d, *logits, *m, *denom, *deg; };

static void run_conv(hipStream_t stream, const float* x_src, const float* x_dst,
                     const float* W_src, const float* W_dst,
                     const float* att_src, const float* att_dst, const float* bias,
                     const int* edges, int E, int n_src, int n_dst,
                     const ConvBufs& wb, float* out) {
    const int* src = edges;
    const int* dst = edges + E;
    compute_wtil<<<1, 128, 0, stream>>>(W_src, att_src, wb.wtil_s);
    compute_wtil<<<1, 128, 0, stream>>>(W_dst, att_dst, wb.wtil_d);
    gemm_wmma<<<(n_src + 127) / 128, 256, 0, stream>>>(x_src, W_src, wb.H, n_src);
    matvec_att<<<(n_src + 7) / 8, 256, 0, stream>>>(x_src, wb.wtil_s, wb.a_src, n_src);
    matvec_att<<<(n_dst + 7) / 8, 256, 0, stream>>>(x_dst, wb.wtil_d, wb.a_dst, n_dst);
    fill_f32<<<(n_dst + 255) / 256, 256, 0, stream>>>(wb.m, -INFINITY, (size_t)n_dst);
    fill_f32<<<(n_dst + 255) / 256, 256, 0, stream>>>(wb.denom, 0.f, (size_t)n_dst);
    fill_f32<<<(n_dst + 255) / 256, 256, 0, stream>>>(wb.deg, 0.f, (size_t)n_dst);
    fill_f32<<<((size_t)n_dst * FC + 255) / 256, 256, 0, stream>>>(out, 0.f, (size_t)n_dst * FC);
    edge_logits_max<<<(E + 255) / 256, 256, 0, stream>>>(src, dst, wb.a_src, wb.a_dst,
                                                         wb.logits, wb.m, wb.deg, E);
    edge_exp_denom<<<(E + 255) / 256, 256, 0, stream>>>(dst, wb.m, wb.logits, wb.denom, E);
    edge_scatter<<<(E * 32 + 255) / 256, 256, 0, stream>>>(src, dst, wb.logits, wb.denom,
                                                           wb.H, out, E);
    finalize_mean_bias<<<(n_dst * FC + 255) / 256, 256, 0, stream>>>(out, wb.deg, bias, n_dst);
}

// Input order (setup_inputs insertion order, nested dicts flattened in order):
// 0 x_user, 1 x_item, 2 ei_u2i[2E], 3 ei_i2u[2E], then per layer (14 tensors):
//   u2i{W_src,W_dst,att_src,att_dst,bias}, i2u{same 5},
//   ln_user{weight,bias}, ln_item{weight,bias}
extern "C" void kernel_launch(void* const* d_in, const int* in_sizes, int n_in,
                              void* d_out, int out_size, void* d_ws, size_t ws_size,
                              hipStream_t stream) {
    const float* x_user0 = (const float*)d_in[0];
    const float* x_item0 = (const float*)d_in[1];
    const int* ei_u2i = (const int*)d_in[2];
    const int* ei_i2u = (const int*)d_in[3];
    int n_user = in_sizes[0] / FC;
    int n_item = in_sizes[1] / FC;
    int E      = in_sizes[2] / 2;
    int maxN   = n_user > n_item ? n_user : n_item;

    auto P = [&](int l, int off) { return (const float*)d_in[4 + l * 14 + off]; };

    float* ws = (float*)d_ws;
    float* xu = ws;  ws += (size_t)n_user * FC;    // layer-1 user features
    float* xi = ws;  ws += (size_t)n_item * FC;    // layer-1 item features
    ConvBufs wb;
    wb.H      = ws;  ws += (size_t)maxN * FC;
    wb.a_src  = ws;  ws += maxN;
    wb.a_dst  = ws;  ws += maxN;
    wb.wtil_s = ws;  ws += FC;
    wb.wtil_d = ws;  ws += FC;
    wb.logits = ws;  ws += E;
    wb.m      = ws;  ws += maxN;
    wb.denom  = ws;  ws += maxN;
    wb.deg    = ws;  ws += maxN;

    float* out_user = (float*)d_out;
    float* out_item = (float*)d_out + (size_t)n_user * FC;

    // ---- layer 0 (reads d_in, writes ws intermediates) ----
    run_conv(stream, x_user0, x_item0, P(0,0), P(0,1), P(0,2), P(0,3), P(0,4),
             ei_u2i, E, n_user, n_item, wb, xi);                    // new_item
    run_conv(stream, x_item0, x_user0, P(0,5), P(0,6), P(0,7), P(0,8), P(0,9),
             ei_i2u, E, n_item, n_user, wb, xu);                    // new_user
    ln_relu<<<(n_user + 7) / 8, 256, 0, stream>>>(xu, P(0,10), P(0,11), n_user);
    ln_relu<<<(n_item + 7) / 8, 256, 0, stream>>>(xi, P(0,12), P(0,13), n_item);

    // ---- layer 1 (reads ws, writes d_out directly) ----
    run_conv(stream, xu, xi, P(1,0), P(1,1), P(1,2), P(1,3), P(1,4),
             ei_u2i, E, n_user, n_item, wb, out_item);
    run_conv(stream, xi, xu, P(1,5), P(1,6), P(1,7), P(1,8), P(1,9),
             ei_i2u, E, n_item, n_user, wb, out_user);
    ln_relu<<<(n_user + 7) / 8, 256, 0, stream>>>(out_user, P(1,10), P(1,11), n_user);
    ln_relu<<<(n_item + 7) / 8, 256, 0, stream>>>(out_item, P(1,12), P(1,13), n_item);
}